// MemoryOnGpu_59837484368075
// MI455X (gfx1250) — compile-verified
//
#include <hip/hip_runtime.h>

typedef __attribute__((ext_vector_type(2))) float v2f;
typedef __attribute__((ext_vector_type(8))) float v8f;

#define NQ 1024
#define ND 8
#define DBN 32768
#define NF 64
#define NB 128                    // buckets == neighbors
#define BSZ 256                   // DBN / NB candidates per bucket
#define OUT_HALF ((size_t)NQ*ND*NB*NF)   // 67,108,864 floats per output tensor

// ---------------------------------------------------------------------------
// Prepass: mask[d][n] = -1e6 if key row sums to zero, else 0.  1 MB in d_ws.
// ---------------------------------------------------------------------------
__global__ __launch_bounds__(256)
void build_zero_mask(const float* __restrict__ key_db, float* __restrict__ mask)
{
    const int id = blockIdx.x * 256 + threadIdx.x;   // 0 .. ND*DBN-1
    const float* row = key_db + (size_t)id * NF;
    float s = 0.0f;
#pragma unroll
    for (int i = 0; i < 16; ++i) {
        float4 v = *(const float4*)(row + 4 * i);
        s += v.x + v.y + v.z + v.w;
    }
    mask[id] = (s == 0.0f) ? -1.0e6f : 0.0f;
}

// ---------------------------------------------------------------------------
// Fused scores-GEMM + per-bucket argmax + gather.
// One wave per (dataset, 32-query tile, 16-bucket tile).
// ---------------------------------------------------------------------------
__global__ __launch_bounds__(32)
void knn_bucket_argmax_gather(const float* __restrict__ query,
                              const float* __restrict__ key_db,
                              const float* __restrict__ value_db,
                              const float* __restrict__ maskbuf,
                              float* __restrict__ out)
{
    const int lane = threadIdx.x;       // 0..31 (wave32)
    const int jt   = blockIdx.x;        // 0..7  : bucket tile (16 buckets)
    const int qt   = blockIdx.y;        // 0..31 : query tile (32 queries)
    const int d    = blockIdx.z;        // 0..7  : dataset
    const int q0   = qt * 32;
    const int j0   = jt * 16;
    const int hi   = lane >> 4;         // lane half: selects K+2 in fp32 WMMA layout
    const int lm   = lane & 15;

    // A tiles: lane lm holds query rows (q0+lm) and (q0+16+lm); per K-step the
    // float2 at k = 4*ks + 2*hi + {0,1}  (16x4 fp32 A VGPR layout)
    v2f a0[16], a1[16];
    const float* qrow0 = query + ((size_t)(q0 + lm) * ND + d) * NF + 2 * hi;
    const float* qrow1 = qrow0 + (size_t)16 * ND * NF;
#pragma unroll
    for (int ks = 0; ks < 16; ++ks) {
        a0[ks] = *(const v2f*)(qrow0 + 4 * ks);
        a1[ks] = *(const v2f*)(qrow1 + 4 * ks);
    }

    float best0[8], best1[8];
    int   bb0[8],   bb1[8];
#pragma unroll
    for (int r = 0; r < 8; ++r) {
        best0[r] = -3.0e38f; best1[r] = -3.0e38f;
        bb0[r] = 0;          bb1[r] = 0;
    }

    // B rows at iteration b: contiguous DB rows 128*b + j0 + lm
    const float* kbase = key_db  + (size_t)d * DBN * NF + (size_t)(j0 + lm) * NF + 2 * hi;
    const float* mbase = maskbuf + (size_t)d * DBN + (j0 + lm);

    for (int b = 0; b < BSZ; ++b) {
        const float* krow  = kbase + (size_t)b * NB * NF;
        const float  maskv = mbase[(size_t)b * NB];      // per-column additive mask

        v2f bv[16];
#pragma unroll
        for (int ks = 0; ks < 16; ++ks)
            bv[ks] = *(const v2f*)(krow + 4 * ks);

        v8f c0 = {}, c1 = {};
#pragma unroll
        for (int ks = 0; ks < 16; ++ks) {
            c0 = __builtin_amdgcn_wmma_f32_16x16x4_f32(
                     false, a0[ks], false, bv[ks], (short)0, c0, false, false);
            c1 = __builtin_amdgcn_wmma_f32_16x16x4_f32(
                     false, a1[ks], false, bv[ks], (short)0, c1, false, false);
        }

        // C layout: vgpr r, lane -> (m = r + 8*hi, j = lm); mask depends only on j.
#pragma unroll
        for (int r = 0; r < 8; ++r) {
            float s0 = c0[r] + maskv;
            if (s0 > best0[r]) { best0[r] = s0; bb0[r] = b; }
            float s1 = c1[r] + maskv;
            if (s1 > best1[r]) { best1[r] = s1; bb1[r] = b; }
        }
    }

    // publish selected DB row index per (m, j); m in [0,32)
    __shared__ int nsel[512];
#pragma unroll
    for (int r = 0; r < 8; ++r) {
        const int m = r + 8 * hi;
        nsel[m * 16 + lm]          = bb0[r] * NB + (j0 + lm);
        nsel[(m + 16) * 16 + lm]   = bb1[r] * NB + (j0 + lm);
    }
    __syncthreads();

    // gather: 512 (q, bucket) pairs; lanes 0-15 copy key rows (float4 each),
    // lanes 16-31 copy value rows.
    const int    fo     = 4 * lm;
    const float* srctab = hi ? value_db : key_db;
    float*       dsttab = hi ? (out + OUT_HALF) : out;
#pragma unroll 8
    for (int pp = 0; pp < 512; ++pp) {
        const int m   = pp >> 4, j = pp & 15;
        const int ndb = nsel[pp];
        const size_t src = ((size_t)d * DBN + ndb) * NF + fo;
        const size_t dst = (((size_t)(q0 + m) * ND + d) * NB + (j0 + j)) * NF + fo;
        *(float4*)(dsttab + dst) = *(const float4*)(srctab + src);
    }
}

extern "C" void kernel_launch(void* const* d_in, const int* in_sizes, int n_in,
                              void* d_out, int out_size, void* d_ws, size_t ws_size,
                              hipStream_t stream) {
    const float* query    = (const float*)d_in[0];
    const float* key_db   = (const float*)d_in[1];
    const float* value_db = (const float*)d_in[2];
    // d_in[3] = num_neighbors scalar (always 128; baked into the kernel)
    float* out  = (float*)d_out;
    float* mask = (float*)d_ws;          // ND*DBN floats = 1 MB

    build_zero_mask<<<(ND * DBN) / 256, 256, 0, stream>>>(key_db, mask);

    dim3 grid(8 /*bucket tiles*/, 32 /*query tiles*/, 8 /*datasets*/);
    knn_bucket_argmax_gather<<<grid, 32, 0, stream>>>(query, key_db, value_db, mask, out);
}